// _ProposalLayer_9371618639963
// MI455X (gfx1250) — compile-verified
//
#include <hip/hip_runtime.h>
#include <stdint.h>

// ---------------- Problem constants (fixed by reference setup_inputs) -------
#define B_      16
#define A_      9
#define H_      128
#define W_      128
#define N_      (H_*W_*A_)      // 147456 anchors per batch
#define NP2_    262144          // next pow2 of N_, bitonic sort size
#define PRE_    12000           // PRE_NMS_TOPN
#define POST_   2000            // POST_NMS_TOPN
#define NWORDS_ 375             // PRE_/32 (exact: 375*32 == 12000)
#define NMS_T   0.7f

typedef __attribute__((ext_vector_type(2))) float v2f;
typedef __attribute__((ext_vector_type(8))) float v8f;

// generate_anchors(base=16, ratios=[.5,1,2], scales=[8,16,32]) evaluated
// exactly (numpy round-half-even): classic Faster-RCNN anchor table.
__device__ __constant__ float kAnc[9][4] = {
  { -84.f,  -40.f,  99.f,  55.f},
  {-176.f,  -88.f, 191.f, 103.f},
  {-360.f, -184.f, 375.f, 199.f},
  { -56.f,  -56.f,  71.f,  71.f},
  {-120.f, -120.f, 135.f, 135.f},
  {-248.f, -248.f, 263.f, 263.f},
  { -36.f,  -84.f,  51.f,  99.f},
  { -80.f, -176.f,  95.f, 191.f},
  {-168.f, -360.f, 183.f, 375.f}
};

// ---------------- 1) key generation ----------------------------------------
// key = (~orderable(score) << 32) | index  -> ascending sort == descending
// score with stable ascending-index tie break (matches stable jnp.argsort).
__global__ void keygen_kernel(const float* __restrict__ scores,
                              uint64_t* __restrict__ keys) {
  int gid = blockIdx.x * blockDim.x + threadIdx.x;   // over B_*NP2_
  if (gid >= B_ * NP2_) return;
  int b = gid >> 18;              // NP2_ == 2^18
  int i = gid & (NP2_ - 1);
  uint64_t key = ~0ull;           // padding sorts last
  if (i < N_) {
    int a = i % A_;
    int s = i / A_;
    int x = s & (W_ - 1);
    int y = s >> 7;
    // scr[b,i] = scores[b, 9+a, y, x]
    float sc = scores[(size_t)b * (2 * A_ * H_ * W_) +
                      (size_t)(A_ + a) * (H_ * W_) + y * W_ + x];
    unsigned u = __float_as_uint(sc);
    u ^= (u >> 31) ? 0xFFFFFFFFu : 0x80000000u;      // monotonic map
    key = ((uint64_t)(~u) << 32) | (unsigned)i;
  }
  keys[(size_t)b * NP2_ + i] = key;
}

// ---------------- 2) bitonic sort pass (in place, L2-resident) --------------
__global__ void bitonic_kernel(uint64_t* __restrict__ keys, int k, int j) {
  int t = blockIdx.x * blockDim.x + threadIdx.x;     // B_ * NP2_/2 pairs
  if (t >= B_ * (NP2_ / 2)) return;
  int b = t / (NP2_ / 2);
  int p = t % (NP2_ / 2);
  int i = 2 * j * (p / j) + (p % j);
  int l = i + j;
  uint64_t* kb = keys + (size_t)b * NP2_;
  bool asc = ((i & k) == 0);
  uint64_t a = kb[i];
  uint64_t c = kb[l];
  if ((a > c) == asc) { kb[i] = c; kb[l] = a; }
}

// ---------------- 3) decode top-12000 proposals -----------------------------
__global__ void gather_kernel(const uint64_t* __restrict__ keys,
                              const float* __restrict__ deltas,
                              const float* __restrict__ im_info,
                              float4* __restrict__ topBoxes,
                              float* __restrict__ topAreas) {
  int gid = blockIdx.x * blockDim.x + threadIdx.x;   // B_*PRE_
  if (gid >= B_ * PRE_) return;
  int b = gid / PRE_;
  int t = gid % PRE_;
  unsigned idx = (unsigned)(keys[(size_t)b * NP2_ + t] & 0xFFFFFFFFu);
  int a = idx % A_;
  int s = idx / A_;
  int x = s & (W_ - 1);
  int y = s >> 7;
  // deltas[b, i, c] = bbox_deltas[b, 4a+c, y, x]
  const float* dp = deltas + (size_t)b * (4 * A_ * H_ * W_) +
                    (size_t)(a * 4) * (H_ * W_) + y * W_ + x;
  float dx = dp[0];
  float dy = dp[H_ * W_];
  float dw = dp[2 * H_ * W_];
  float dh = dp[3 * H_ * W_];
  float sxf = 16.f * (float)x;
  float syf = 16.f * (float)y;
  float ax1 = kAnc[a][0] + sxf, ay1 = kAnc[a][1] + syf;
  float ax2 = kAnc[a][2] + sxf, ay2 = kAnc[a][3] + syf;
  float aw = ax2 - ax1 + 1.f, ah = ay2 - ay1 + 1.f;
  float acx = ax1 + 0.5f * aw, acy = ay1 + 0.5f * ah;     // as in reference
  float pcx = dx * aw + acx,   pcy = dy * ah + acy;
  float pw = expf(dw) * aw,    ph = expf(dh) * ah;
  float maxx = im_info[b * 3 + 1] - 1.f;
  float maxy = im_info[b * 3 + 0] - 1.f;
  float x1 = fminf(fmaxf(pcx - 0.5f * pw, 0.f), maxx);
  float y1 = fminf(fmaxf(pcy - 0.5f * ph, 0.f), maxy);
  float x2 = fminf(fmaxf(pcx + 0.5f * pw, 0.f), maxx);
  float y2 = fminf(fmaxf(pcy + 0.5f * ph, 0.f), maxy);
  topBoxes[gid] = make_float4(x1, y1, x2, y2);
  topAreas[gid] = (x2 - x1 + 1.f) * (y2 - y1 + 1.f);
}

// ---------------- 4) NMS suppression bitmask: WMMA area-sum + ballot --------
// One wave32 computes a 16-row x 32-col block of the 12000x12000 IoU>thresh
// bitmask. The scaled denominator term 0.7*(area_i + area_j) is an outer
// product computed on the matrix pipe: A(16x4)=[0.7*area_m, 0.7, 0, 0] rows,
// B(4x16)=[1, area_n, 0, 0] cols -> D[m][n]=0.7*(area_m+area_n) via
// V_WMMA_F32_16X16X4_F32. The per-pair predicate
//   iou > 0.7  <=>  inter/(S - inter) > 0.7  <=>  1.7*inter > 0.7*S   (S>0)
// then needs only v_mul + v_cmp (no IEEE divide sequence).
__global__ void __launch_bounds__(256)
mask_kernel(const float4* __restrict__ boxes,
            const float* __restrict__ areas,
            unsigned* __restrict__ mask) {
  const int ROWT = PRE_ / 16;    // 750 row tiles
  const int COLT = NWORDS_;      // 375 col words (32 cols each)
  int wave = blockIdx.x * (blockDim.x >> 5) + (threadIdx.x >> 5);
  if (wave >= ROWT * COLT) return;   // wave-uniform exit; EXEC stays full
  int ct = wave % COLT;
  int rt = wave / COLT;
  // Dead-work skip: words entirely below the diagonal (all 32 cols < every
  // row in this 16-row tile) only ever OR bits at j < i into the scan's
  // suppressed bitmap, which are consumed before they could be read again.
  // Their (stale) contents are provably never observed -> skip, don't zero.
  if (ct * 32 + 31 < rt * 16) return;  // wave-uniform
  int lane = threadIdx.x & 31;
  int nc = lane & 15;
  bool lo = lane < 16;

  // per-lane column boxes for the two 16-wide tiles
  float4 cb0 = boxes[ct * 32 + nc];
  float4 cb1 = boxes[ct * 32 + 16 + nc];

  // A 16x4 layout: lanes0-15 {VGPR0=K0, VGPR1=K1}; lanes16-31 {K2,K3}=0
  v2f Aop;
  Aop[0] = lo ? NMS_T * areas[rt * 16 + nc] : 0.f;   // K0 = 0.7*area_m
  Aop[1] = lo ? NMS_T : 0.f;                         // K1 = 0.7
  // B 4x16 layout mirrored: lanes hold column n = lane&15
  v2f B0, B1;
  B0[0] = lo ? 1.f : 0.f;                    // K0 row = 1
  B0[1] = lo ? areas[ct * 32 + nc] : 0.f;    // K1 row = area_n
  B1[0] = B0[0];
  B1[1] = lo ? areas[ct * 32 + 16 + nc] : 0.f;

  v8f c0 = {};
  v8f c1 = {};
  // D = A x B (+0): D[m][n] = 0.7*(area_m + area_n)
  c0 = __builtin_amdgcn_wmma_f32_16x16x4_f32(false, Aop, false, B0,
                                             (short)0, c0, false, false);
  c1 = __builtin_amdgcn_wmma_f32_16x16x4_f32(false, Aop, false, B1,
                                             (short)0, c1, false, false);

  int rbase = rt * 16 + (lo ? 0 : 8);   // C/D layout: VGPR r -> M=r / M=r+8
#pragma unroll
  for (int r = 0; r < 8; ++r) {
    float4 rb = boxes[rbase + r];       // broadcast load (2 addrs per wave)
    // tile 0 (cols ct*32 .. +15)
    float xx1 = fmaxf(rb.x, cb0.x), yy1 = fmaxf(rb.y, cb0.y);
    float xx2 = fminf(rb.z, cb0.z), yy2 = fminf(rb.w, cb0.w);
    float in0 = fmaxf(xx2 - xx1 + 1.f, 0.f) * fmaxf(yy2 - yy1 + 1.f, 0.f);
    unsigned m0 =
        __builtin_amdgcn_ballot_w32((1.f + NMS_T) * in0 > c0[r]);
    // tile 1 (cols ct*32+16 .. +31)
    xx1 = fmaxf(rb.x, cb1.x); yy1 = fmaxf(rb.y, cb1.y);
    xx2 = fminf(rb.z, cb1.z); yy2 = fminf(rb.w, cb1.w);
    float in1 = fmaxf(xx2 - xx1 + 1.f, 0.f) * fmaxf(yy2 - yy1 + 1.f, 0.f);
    unsigned m1 =
        __builtin_amdgcn_ballot_w32((1.f + NMS_T) * in1 > c1[r]);
    // ballot bits: [15:0] -> row M=r cols0-15, [31:16] -> row M=r+8 cols0-15
    if (lane == 0) {
      mask[(size_t)(rt * 16 + r) * NWORDS_ + ct] =
          (m0 & 0xFFFFu) | ((m1 & 0xFFFFu) << 16);
      mask[(size_t)(rt * 16 + 8 + r) * NWORDS_ + ct] =
          (m0 >> 16) | (m1 & 0xFFFF0000u);
    }
  }
}

// ---------------- 5) sequential greedy scan (single wave, LDS bitmap) -------
__global__ void __launch_bounds__(32)
scan_kernel(const float4* __restrict__ boxes,
            const unsigned* __restrict__ mask,
            float* __restrict__ out, float bval) {
  __shared__ unsigned supp[NWORDS_];
  int lane = threadIdx.x;
  for (int w = lane; w < NWORDS_; w += 32) supp[w] = 0u;
  // init output rows: {b, 0,0,0,0} (reference zeroes invalid keeps)
  for (int s = lane; s < POST_; s += 32) {
    out[s * 5 + 0] = bval;
    out[s * 5 + 1] = 0.f; out[s * 5 + 2] = 0.f;
    out[s * 5 + 3] = 0.f; out[s * 5 + 4] = 0.f;
  }
  __syncthreads();
  int k = 0;
  for (int i = 0; i < PRE_ && k < POST_; ++i) {
    if (i + 1 < PRE_)   // speculative prefetch of next mask row (gfx1250 path)
      __builtin_prefetch(mask + (size_t)(i + 1) * NWORDS_ + lane, 0, 1);
    unsigned w = supp[i >> 5];          // all lanes: same address -> uniform
    if (!((w >> (i & 31)) & 1u)) {      // box i survives -> keep it
      if (lane == 0) {
        float4 bx = boxes[i];
        out[k * 5 + 1] = bx.x; out[k * 5 + 2] = bx.y;
        out[k * 5 + 3] = bx.z; out[k * 5 + 4] = bx.w;
      }
      const unsigned* row = mask + (size_t)i * NWORDS_;
      for (int t = lane; t < NWORDS_; t += 32) supp[t] |= row[t];
      ++k;
    }
    __syncthreads();   // single-wave WG: lowers to S_NOP + DS waits
  }
}

// ---------------- launcher --------------------------------------------------
extern "C" void kernel_launch(void* const* d_in, const int* in_sizes, int n_in,
                              void* d_out, int out_size, void* d_ws,
                              size_t ws_size, hipStream_t stream) {
  const float* scores  = (const float*)d_in[0];
  const float* deltas  = (const float*)d_in[1];
  const float* im_info = (const float*)d_in[2];
  float* out = (float*)d_out;

  // workspace layout (~53 MB total)
  char* ws = (char*)d_ws;
  uint64_t* keys   = (uint64_t*)ws;                                   // 32 MB
  float4* topBoxes = (float4*)(ws + (size_t)B_ * NP2_ * 8);           // 3 MB
  float* topAreas  = (float*)((char*)topBoxes +
                              (size_t)B_ * PRE_ * sizeof(float4));    // .75 MB
  unsigned* mask   = (unsigned*)((char*)topAreas +
                                 (size_t)B_ * PRE_ * sizeof(float));  // 18 MB

  {
    int n = B_ * NP2_;
    keygen_kernel<<<(n + 255) / 256, 256, 0, stream>>>(scores, keys);
  }
  // full ascending bitonic sort per batch (keys array fits in 192MB L2)
  for (int k = 2; k <= NP2_; k <<= 1)
    for (int j = k >> 1; j > 0; j >>= 1) {
      int n = B_ * (NP2_ / 2);
      bitonic_kernel<<<(n + 255) / 256, 256, 0, stream>>>(keys, k, j);
    }
  {
    int n = B_ * PRE_;
    gather_kernel<<<(n + 255) / 256, 256, 0, stream>>>(keys, deltas, im_info,
                                                       topBoxes, topAreas);
  }
  const int tiles = (PRE_ / 16) * NWORDS_;   // 750 x 375 wave tiles
  for (int b = 0; b < B_; ++b) {
    mask_kernel<<<(tiles + 7) / 8, 256, 0, stream>>>(
        topBoxes + (size_t)b * PRE_, topAreas + (size_t)b * PRE_, mask);
    scan_kernel<<<1, 32, 0, stream>>>(topBoxes + (size_t)b * PRE_, mask,
                                      out + (size_t)b * POST_ * 5, (float)b);
  }
}